// PagedAttention_37349035606361
// MI455X (gfx1250) — compile-verified
//
#include <hip/hip_runtime.h>
#include <math.h>

typedef float v2f __attribute__((ext_vector_type(2)));
typedef float v8f __attribute__((ext_vector_type(8)));

static constexpr int kB = 8;
static constexpr int kH = 32;
static constexpr int kS = 8192;
static constexpr int kD = 128;
static constexpr int kBH = kB * kH;            // 256
static constexpr int kHD = kH * kD;            // 4096
static constexpr int kNSplit = 8;
static constexpr int kSSplit = kS / kNSplit;   // 1024
static constexpr int kWaves = 8;
static constexpr int kSWave = kSSplit / kWaves; // 128
static constexpr int kChunks = kSWave / 16;     // 8
static constexpr int kPartStride = 2 + kD;      // m, l, o[128]
static constexpr size_t kPartFloats = (size_t)kBH * kNSplit * kPartStride;
static constexpr float kInvSqrtD = 0.08838834764831845f; // 1/sqrt(128)

// -------------------------------------------------------------------------
// Kernel 1: flash-decoding partial attention per (b*h, split).
// WMMA f32 16x16x4 computes 16 scores at a time (q replicated across M).
// -------------------------------------------------------------------------
__global__ __launch_bounds__(256) void attn_partial_kernel(
    const float* __restrict__ q,     // [BH, D]
    const float* __restrict__ Kc,    // [BH, S, D]
    const float* __restrict__ Vc,    // [BH, S, D]
    float* __restrict__ part)        // [BH, NSPLIT, 2+D]
{
    __shared__ float q_s[kD];
    __shared__ float m_s[kWaves];
    __shared__ float l_s[kWaves];
    __shared__ float o_s[kWaves][kD];

    const int bh    = blockIdx.x;
    const int split = blockIdx.y;
    const int t     = threadIdx.x;
    const int wave  = t >> 5;
    const int lane  = t & 31;
    const int lo    = lane & 15;   // B-matrix column (s within chunk)
    const int hi    = lane >> 4;   // k-pair selector

    if (t < kD) q_s[t] = q[(size_t)bh * kD + t];
    __syncthreads();

    const size_t bh_base = (size_t)bh * kS * kD;
    const int s_start = split * kSSplit + wave * kSWave;

    float m = -INFINITY, l = 0.f;
    float o0 = 0.f, o1 = 0.f, o2 = 0.f, o3 = 0.f;   // lane owns d = 4*lane..+3

    const float* qk = q_s + hi * 2;

    for (int c = 0; c < kChunks; ++c) {
        const int s_base = s_start + c * 16;

        // ---- scores for 16 s-positions via 32 x wmma_f32_16x16x4_f32 ----
        v8f acc = {0.f, 0.f, 0.f, 0.f, 0.f, 0.f, 0.f, 0.f};
        const float* krow = Kc + bh_base + (size_t)(s_base + lo) * kD + hi * 2;
        #pragma unroll
        for (int dstep = 0; dstep < 32; ++dstep) {
            v2f a  = *(const v2f*)(qk + 4 * dstep);     // A: q replicated rows
            v2f bm = *(const v2f*)(krow + 4 * dstep);   // B: K[s_base+lo, d..]
            acc = __builtin_amdgcn_wmma_f32_16x16x4_f32(
                false, a, false, bm, (short)0, acc, false, false);
        }
        // C VGPR0: lane L holds score[s_base + (L&15)] (rows identical)
        float score = acc[0] * kInvSqrtD;

        // ---- online softmax update (reduce within 16-lane group) ----
        float cm = score;
        cm = fmaxf(cm, __shfl_xor(cm, 1));
        cm = fmaxf(cm, __shfl_xor(cm, 2));
        cm = fmaxf(cm, __shfl_xor(cm, 4));
        cm = fmaxf(cm, __shfl_xor(cm, 8));
        float m_new = fmaxf(m, cm);
        float scale = __expf(m - m_new);
        float p     = __expf(score - m_new);
        float ps = p;
        ps += __shfl_xor(ps, 1);
        ps += __shfl_xor(ps, 2);
        ps += __shfl_xor(ps, 4);
        ps += __shfl_xor(ps, 8);
        l = l * scale + ps;
        o0 *= scale; o1 *= scale; o2 *= scale; o3 *= scale;
        m = m_new;

        // ---- P*V: broadcast p_j, stream full V rows (coalesced b128) ----
        const float* vrow = Vc + bh_base + (size_t)s_base * kD + 4 * lane;
        #pragma unroll
        for (int j = 0; j < 16; ++j) {
            float pj = __shfl(p, j);   // lane j holds p for s_base+j
            float4 vv = *(const float4*)(vrow + (size_t)j * kD);
            o0 += pj * vv.x; o1 += pj * vv.y; o2 += pj * vv.z; o3 += pj * vv.w;
        }
    }

    // ---- combine the 8 wave partials of this block ----
    o_s[wave][4 * lane + 0] = o0;
    o_s[wave][4 * lane + 1] = o1;
    o_s[wave][4 * lane + 2] = o2;
    o_s[wave][4 * lane + 3] = o3;
    if (lane == 0) { m_s[wave] = m; l_s[wave] = l; }
    __syncthreads();

    if (t < kD) {
        float mm = m_s[0];
        #pragma unroll
        for (int w = 1; w < kWaves; ++w) mm = fmaxf(mm, m_s[w]);
        float ll = 0.f, oo = 0.f;
        #pragma unroll
        for (int w = 0; w < kWaves; ++w) {
            float e = __expf(m_s[w] - mm);
            ll += l_s[w] * e;
            oo += o_s[w][t] * e;
        }
        float* pb = part + ((size_t)bh * kNSplit + split) * kPartStride;
        if (t == 0) { pb[0] = mm; pb[1] = ll; }
        pb[2 + t] = oo;
    }
}

// -------------------------------------------------------------------------
// Kernel 2: combine the NSPLIT partials -> attn[b*4096 + h*128 + d]
// -------------------------------------------------------------------------
__global__ __launch_bounds__(128) void attn_combine_kernel(
    const float* __restrict__ part, float* __restrict__ attn)
{
    const int bh = blockIdx.x;
    const int d  = threadIdx.x;
    const float* pb = part + (size_t)bh * kNSplit * kPartStride;

    float mm = pb[0];
    #pragma unroll
    for (int sp = 1; sp < kNSplit; ++sp)
        mm = fmaxf(mm, pb[(size_t)sp * kPartStride]);

    float ll = 0.f, oo = 0.f;
    #pragma unroll
    for (int sp = 0; sp < kNSplit; ++sp) {
        const float* ps = pb + (size_t)sp * kPartStride;
        float e = __expf(ps[0] - mm);
        ll += ps[1] * e;
        oo += ps[2 + d] * e;
    }
    attn[(size_t)bh * kD + d] = oo / ll;   // == [b*4096 + h*128 + d]
}

// -------------------------------------------------------------------------
// Kernel 3: out = attn[8,4096] @ W^T + b via WMMA f32 16x16x4.
// M padded to 16 (rows 8..15 alias rows 0..7; only rows 0..7 stored).
// 8 waves/block, each wave owns one 16-column tile of W.
// -------------------------------------------------------------------------
__global__ __launch_bounds__(256) void outproj_kernel(
    const float* __restrict__ attn,  // [8, 4096]
    const float* __restrict__ W,     // [4096, 4096] row-major (torch Linear)
    const float* __restrict__ bias,  // [4096]
    float* __restrict__ out)         // [8, 4096]
{
    const int t    = threadIdx.x;
    const int wave = t >> 5;
    const int lane = t & 31;
    const int lo   = lane & 15;
    const int hi   = lane >> 4;
    const int tile = blockIdx.x * 8 + wave;   // 0..255
    const int j0   = tile * 16;

    const float* wrow = W    + (size_t)(j0 + lo) * kHD + hi * 2;     // B: W[j0+lo, k..]
    const float* arow = attn + (size_t)(lo & 7) * kHD + hi * 2;      // A: attn[b, k..]

    v8f acc = {0.f, 0.f, 0.f, 0.f, 0.f, 0.f, 0.f, 0.f};
    #pragma unroll 8
    for (int kstep = 0; kstep < kHD / 4; ++kstep) {
        v2f a  = *(const v2f*)(arow + 4 * kstep);
        v2f bm = *(const v2f*)(wrow + 4 * kstep);
        acc = __builtin_amdgcn_wmma_f32_16x16x4_f32(
            false, a, false, bm, (short)0, acc, false, false);
    }

    if (lane < 16) {
        float bj = bias[j0 + lane];
        #pragma unroll
        for (int r = 0; r < kB; ++r)             // C VGPR r = row b=r, col=lane
            out[(size_t)r * kHD + j0 + lane] = acc[r] + bj;
    }
}

// -------------------------------------------------------------------------
extern "C" void kernel_launch(void* const* d_in, const int* in_sizes, int n_in,
                              void* d_out, int out_size, void* d_ws, size_t ws_size,
                              hipStream_t stream) {
    const float* q    = (const float*)d_in[0];   // [8,32,1,128]
    const float* Kc   = (const float*)d_in[1];   // [8,32,8192,128]
    const float* Vc   = (const float*)d_in[2];   // [8,32,8192,128]
    const float* Wout = (const float*)d_in[3];   // [4096,4096]
    const float* bout = (const float*)d_in[4];   // [4096]
    float* out = (float*)d_out;                  // [8,1,4096]

    float* ws   = (float*)d_ws;
    float* part = ws;                            // kPartFloats floats
    float* attn = ws + kPartFloats;              // 8*4096 floats

    dim3 g1(kBH, kNSplit);
    attn_partial_kernel<<<g1, 256, 0, stream>>>(q, Kc, Vc, part);
    attn_combine_kernel<<<kBH, 128, 0, stream>>>(part, attn);
    outproj_kernel<<<32, 256, 0, stream>>>(attn, Wout, bout, out);
}